// SketchEncoder_33139967656329
// MI455X (gfx1250) — compile-verified
//
#include <hip/hip_runtime.h>
#include <hip/hip_bf16.h>

// ---------------- problem constants ----------------
#define N_NODES   50000
#define NFEAT     12
#define HID       256
#define NLAYERS   4
#define HEADS     4
#define PHEAD     64
#define NEDGES    800000
#define ETOT      (NEDGES + N_NODES)   // self-loops appended
#define OUTD      512
#define LN_EPS    1e-5f
#define NEG_SLOPE 0.2f

typedef float v2f __attribute__((ext_vector_type(2)));
typedef float v8f __attribute__((ext_vector_type(8)));

// ---------------- helpers ----------------
__device__ __forceinline__ unsigned enc_f32(float f) {
  int i = __float_as_int(f);
  return (i >= 0) ? ((unsigned)i | 0x80000000u) : ~(unsigned)i;
}
__device__ __forceinline__ float dec_f32(unsigned u) {
  int i = (u & 0x80000000u) ? (int)(u & 0x7fffffffu) : (int)(~u);
  return __int_as_float(i);
}
#define ENC_NEG_INF 0x007FFFFFu   // enc_f32(-inf)

__device__ __forceinline__ float wave_allsum(float v) {
  #pragma unroll
  for (int off = 16; off > 0; off >>= 1) v += __shfl_xor(v, off, 32);
  return v;
}

// ---------------- input projection: x = nf @ in_W + in_b ----------------
__global__ void k_input_proj(const float* __restrict__ nf,
                             const float* __restrict__ W,
                             const float* __restrict__ b,
                             float* __restrict__ x) {
  int tid = blockIdx.x * 256 + threadIdx.x;          // grid covers N*HID exactly
  int n = tid >> 8, c = tid & 255;
  float acc = b[c];
  #pragma unroll
  for (int k = 0; k < NFEAT; ++k)
    acc += nf[n * NFEAT + k] * W[k * HID + c];
  x[tid] = acc;
}

// ---------------- fp32 WMMA GEMM: Y[N,256] = X[N,256] * B (+ bias) --------
// Compile-time strides: B element (k,n) at W[k*SK + n*SJ].
//   SK=HID, SJ=1  : Y = X @ W
//   SK=1,  SJ=HID : Y = X @ W^T   (B pair contiguous -> b64 loads)
// Block = 128 threads = 4 waves = one 16-row m-tile.
// Each wave owns 4 n-tiles (cols 64*wave + {0,16,32,48}) -> A frag reused 4x,
// 4 independent WMMA accumulator chains per K-step.
template <int SK, int SJ>
__global__ void k_gemm_wmma(const float* __restrict__ X,
                            const float* __restrict__ W,
                            const float* __restrict__ bias,
                            float* __restrict__ Y) {
  const int lane = threadIdx.x & 31;
  const int wave = threadIdx.x >> 5;                 // 0..3
  const int mT   = blockIdx.x;                       // 0..3124
  const int hs   = lane >> 4;                        // K-half select (0/1)
  const int l15  = lane & 15;
  const float* xrow = X + (size_t)(mT * 16 + l15) * HID;
  __builtin_prefetch(xrow + 128, 0, 1);              // global_prefetch_b8
  const int n0 = wave * 64 + l15;                    // column of n-tile 0

  v8f acc[4] = {{}, {}, {}, {}};
  #pragma unroll 8
  for (int k0 = 0; k0 < HID; k0 += 4) {
    const int ka = k0 + hs * 2;
    v2f a; a.x = xrow[ka]; a.y = xrow[ka + 1];
    #pragma unroll
    for (int t = 0; t < 4; ++t) {
      const int n = n0 + 16 * t;
      v2f b; b.x = W[(size_t)ka * SK + (size_t)n * SJ];
             b.y = W[(size_t)(ka + 1) * SK + (size_t)n * SJ];
      acc[t] = __builtin_amdgcn_wmma_f32_16x16x4_f32(false, a, false, b,
                                                     (short)0, acc[t],
                                                     false, false);
    }
  }
  const int rowBase = mT * 16 + hs * 8;              // VGPR r -> row r / r+8
  #pragma unroll
  for (int t = 0; t < 4; ++t) {
    const int col = n0 + 16 * t;
    const float bv = bias ? bias[col] : 0.0f;
    #pragma unroll
    for (int r = 0; r < 8; ++r)
      Y[(size_t)(rowBase + r) * HID + col] = acc[t][r] + bv;
  }
}

// ---------------- per-node attention logits ----------------
__global__ void k_alpha(const float* __restrict__ xl,
                        const float* __restrict__ a_s,
                        const float* __restrict__ a_d,
                        float* __restrict__ asrc,
                        float* __restrict__ adst) {
  int tid = blockIdx.x * 256 + threadIdx.x;
  if (tid >= N_NODES * HEADS) return;
  int n = tid >> 2, h = tid & 3;
  const float* v  = xl + (size_t)n * HID + h * PHEAD;
  const float* ws = a_s + h * PHEAD;
  const float* wd = a_d + h * PHEAD;
  float s = 0.f, d = 0.f;
  #pragma unroll 8
  for (int i = 0; i < PHEAD; ++i) { s += v[i] * ws[i]; d += v[i] * wd[i]; }
  asrc[tid] = s; adst[tid] = d;
}

// ---------------- per-layer init of segment buffers ----------------
__global__ void k_init(float* __restrict__ agg, unsigned* __restrict__ menc,
                       float* __restrict__ z) {
  int tid = blockIdx.x * 256 + threadIdx.x;          // grid covers N*HID
  agg[tid] = 0.0f;
  if (tid < N_NODES * HEADS) { menc[tid] = ENC_NEG_INF; z[tid] = 0.0f; }
}

// ---------------- edge pass 1: leaky logits + segment max ----------------
__global__ void k_edge_max(const int* __restrict__ ei,
                           const float* __restrict__ asrc,
                           const float* __restrict__ adst,
                           unsigned* __restrict__ menc,
                           float* __restrict__ ebuf) {
  int e = blockIdx.x * 256 + threadIdx.x;
  if (e >= ETOT) return;
  int s, d;
  if (e < NEDGES) { s = ei[e]; d = ei[NEDGES + e]; } else { s = d = e - NEDGES; }
  #pragma unroll
  for (int h = 0; h < HEADS; ++h) {
    float ev = asrc[s * HEADS + h] + adst[d * HEADS + h];
    ev = (ev > 0.f) ? ev : NEG_SLOPE * ev;
    ebuf[(size_t)e * HEADS + h] = ev;
    atomicMax(&menc[d * HEADS + h], enc_f32(ev));
  }
}

// ---------------- edge pass 2: exp + segment sum ----------------
__global__ void k_edge_expsum(const int* __restrict__ ei,
                              const unsigned* __restrict__ menc,
                              float* __restrict__ ebuf,
                              float* __restrict__ z) {
  int e = blockIdx.x * 256 + threadIdx.x;
  if (e >= ETOT) return;
  int d = (e < NEDGES) ? ei[NEDGES + e] : (e - NEDGES);
  #pragma unroll
  for (int h = 0; h < HEADS; ++h) {
    float ex = __expf(ebuf[(size_t)e * HEADS + h] - dec_f32(menc[d * HEADS + h]));
    ebuf[(size_t)e * HEADS + h] = ex;
    atomicAdd(&z[d * HEADS + h], ex);
  }
}

// ---------------- edge pass 3: weighted scatter (one wave per edge) -------
// Lane owns channels c = lane*8 .. lane*8+7 (contiguous, single head since
// 8 | PHEAD): two b128 loads of xl[src], one weight per lane, 8 coalesced
// f32 atomics into agg[dst].
__global__ void k_edge_scatter(const int* __restrict__ ei,
                               const float* __restrict__ xl,
                               const float* __restrict__ ebuf,
                               const float* __restrict__ z,
                               float* __restrict__ agg) {
  int lane = threadIdx.x & 31;
  int e = blockIdx.x * 8 + (threadIdx.x >> 5);
  if (e >= ETOT) return;
  int s, d;
  if (e < NEDGES) { s = ei[e]; d = ei[NEDGES + e]; } else { s = d = e - NEDGES; }
  const int h = lane >> 3;                           // head of this lane's 8 chans
  const float w = ebuf[(size_t)e * HEADS + h] / z[d * HEADS + h];
  const float4* xs4 = (const float4*)(xl + (size_t)s * HID);
  float4 u0 = xs4[lane * 2 + 0];
  float4 u1 = xs4[lane * 2 + 1];
  float* ad = agg + (size_t)d * HID + lane * 8;
  atomicAdd(&ad[0], u0.x * w); atomicAdd(&ad[1], u0.y * w);
  atomicAdd(&ad[2], u0.z * w); atomicAdd(&ad[3], u0.w * w);
  atomicAdd(&ad[4], u1.x * w); atomicAdd(&ad[5], u1.y * w);
  atomicAdd(&ad[6], u1.z * w); atomicAdd(&ad[7], u1.w * w);
}

// ---------------- LN + ReLU (one wave per node) ----------------
__global__ void k_ln_relu(const float* __restrict__ agg,
                          const float* __restrict__ xres,
                          const float* __restrict__ gb,
                          const float* __restrict__ g,
                          const float* __restrict__ b,
                          float* __restrict__ x) {
  int lane = threadIdx.x & 31;
  int n = blockIdx.x * 8 + (threadIdx.x >> 5);       // grid covers N exactly
  const float* ar = agg + (size_t)n * HID;
  const float* rr = xres + (size_t)n * HID;
  float v[8], s = 0.f;
  #pragma unroll
  for (int r = 0; r < 8; ++r) {
    int c = lane + 32 * r;
    v[r] = ar[c] + gb[c] + rr[c];
    s += v[r];
  }
  float mu = wave_allsum(s) * (1.0f / HID);
  float q = 0.f;
  #pragma unroll
  for (int r = 0; r < 8; ++r) { float t = v[r] - mu; q += t * t; }
  float rs = rsqrtf(wave_allsum(q) * (1.0f / HID) + LN_EPS);
  float* xr = x + (size_t)n * HID;
  #pragma unroll
  for (int r = 0; r < 8; ++r) {
    int c = lane + 32 * r;
    float y = (v[r] - mu) * rs * g[c] + b[c];
    xr[c] = fmaxf(y, 0.0f);
  }
}

// ---------------- global mean ----------------
__global__ void k_mean_partial(const float* __restrict__ x, float* __restrict__ p) {
  int c = threadIdx.x;
  int n0 = blockIdx.x * 200;                          // 250 blocks * 200 nodes
  float acc = 0.f;
  for (int i = 0; i < 200; ++i) acc += x[(size_t)(n0 + i) * HID + c];
  p[blockIdx.x * HID + c] = acc;
}
__global__ void k_mean_final(const float* __restrict__ p, float* __restrict__ m) {
  int c = threadIdx.x;
  float acc = 0.f;
  for (int i = 0; i < 250; ++i) acc += p[i * HID + c];
  m[c] = acc * (1.0f / N_NODES);
}

// ---------------- output projection (one wave per output) ----------------
__global__ void k_out_proj(const float* __restrict__ m,
                           const float* __restrict__ W,
                           const float* __restrict__ b,
                           float* __restrict__ out) {
  int lane = threadIdx.x & 31;
  int o = blockIdx.x * 8 + (threadIdx.x >> 5);        // 64 blocks -> 512 outputs
  const float* wr = W + (size_t)o * HID;
  float s = 0.f;
  #pragma unroll
  for (int r = 0; r < 8; ++r) { int c = lane + 32 * r; s += wr[c] * m[c]; }
  s = wave_allsum(s);
  if (lane == 0) out[o] = s + b[o];
}

// ---------------- host launch ----------------
extern "C" void kernel_launch(void* const* d_in, const int* in_sizes, int n_in,
                              void* d_out, int out_size, void* d_ws, size_t ws_size,
                              hipStream_t stream) {
  const float* nf      = (const float*)d_in[0];
  const int*   ei      = (const int*)  d_in[1];
  const float* in_W    = (const float*)d_in[2];
  const float* in_b    = (const float*)d_in[3];
  const float* gat_W   = (const float*)d_in[4];
  const float* att_src = (const float*)d_in[5];
  const float* att_dst = (const float*)d_in[6];
  const float* gat_b   = (const float*)d_in[7];
  const float* res_W   = (const float*)d_in[8];
  const float* res_b   = (const float*)d_in[9];
  const float* ln_g    = (const float*)d_in[10];
  const float* ln_b    = (const float*)d_in[11];
  const float* out_W   = (const float*)d_in[12];
  const float* out_b   = (const float*)d_in[13];
  float* out = (float*)d_out;

  // workspace carve (floats)
  const size_t NH = (size_t)N_NODES * HID;
  float*    x     = (float*)d_ws;
  float*    xl    = x     + NH;
  float*    xres  = xl    + NH;
  float*    agg   = xres  + NH;
  float*    asrc  = agg   + NH;
  float*    adst  = asrc  + (size_t)N_NODES * HEADS;
  unsigned* menc  = (unsigned*)(adst + (size_t)N_NODES * HEADS);
  float*    z     = (float*)(menc + (size_t)N_NODES * HEADS);
  float*    ebuf  = z     + (size_t)N_NODES * HEADS;
  float*    part  = ebuf  + (size_t)ETOT * HEADS;
  float*    meanv = part  + 250 * HID;

  const dim3 B256(256);
  const dim3 B128(128);
  const int gNH   = (int)(NH / 256);                 // 50000
  const int gGemm = N_NODES / 16;                    // 3125 (one m-tile / block)
  const int gA    = (N_NODES * HEADS + 255) / 256;   // 782
  const int gE    = (ETOT + 255) / 256;              // 3321
  const int gSc   = (ETOT + 7) / 8;                  // 106250
  const int gLN   = N_NODES / 8;                     // 6250

  k_input_proj<<<gNH, B256, 0, stream>>>(nf, in_W, in_b, x);

  for (int l = 0; l < NLAYERS; ++l) {
    const float* gW = gat_W + (size_t)l * HID * HID;
    const float* rW = res_W + (size_t)l * HID * HID;
    // x_res = x @ res_W[l]^T + res_b[l]   (B[k][n] = rW[n*HID + k])
    k_gemm_wmma<1, HID><<<gGemm, B128, 0, stream>>>(x, rW, res_b + l * HID, xres);
    // xl = x @ gat_W[l]                   (B[k][n] = gW[k*HID + n])
    k_gemm_wmma<HID, 1><<<gGemm, B128, 0, stream>>>(x, gW, nullptr, xl);
    k_alpha<<<gA, B256, 0, stream>>>(xl, att_src + l * HEADS * PHEAD,
                                     att_dst + l * HEADS * PHEAD, asrc, adst);
    k_init<<<gNH, B256, 0, stream>>>(agg, menc, z);
    k_edge_max<<<gE, B256, 0, stream>>>(ei, asrc, adst, menc, ebuf);
    k_edge_expsum<<<gE, B256, 0, stream>>>(ei, menc, ebuf, z);
    k_edge_scatter<<<gSc, B256, 0, stream>>>(ei, xl, ebuf, z, agg);
    k_ln_relu<<<gLN, B256, 0, stream>>>(agg, xres, gat_b + l * HID,
                                        ln_g + l * HID, ln_b + l * HID, x);
  }

  k_mean_partial<<<250, B256, 0, stream>>>(x, part);
  k_mean_final<<<1, B256, 0, stream>>>(part, meanv);
  k_out_proj<<<OUTD / 8, B256, 0, stream>>>(meanv, out_W, out_b, out);
}